// MambaBlock_84576495993556
// MI455X (gfx1250) — compile-verified
//
#include <hip/hip_runtime.h>
#include <hip/hip_bf16.h>
#include <math.h>

// Problem dimensions (fixed by the reference)
#define DM   768
#define DI   1536
#define DTR  48
#define DS   16
#define DC   4
#define BB   2
#define SL   2048
#define ROWS (BB * SL)        // 4096
#define XPN  (DTR + 2 * DS)   // 80
#define CH   16               // scan chunks per sequence
#define CL   (SL / CH)        // 128 steps per chunk

typedef __attribute__((ext_vector_type(2))) float v2f;
typedef __attribute__((ext_vector_type(8))) float v8f;
typedef __attribute__((ext_vector_type(4))) int   v4i;

enum { EP_PLAIN = 0, EP_SPLIT = 1, EP_SOFTPLUS = 2 };

// CDNA5 async global->LDS path (guarded; falls back to sync staging)
#if __has_builtin(__builtin_amdgcn_global_load_async_to_lds_b128) && \
    __has_builtin(__builtin_amdgcn_s_wait_asynccnt)
#define USE_ASYNC_LDS 1
#else
#define USE_ASYNC_LDS 0
#endif

// exact parameter types per hipcc diagnostic: global src is (AS1) v4i*, LDS dst is (AS3) v4i*
typedef __attribute__((address_space(1))) v4i* gv4i_p;
typedef __attribute__((address_space(3))) v4i* lv4i_p;

__device__ __forceinline__ float silu_f(float v) {
    return v / (1.0f + __expf(-v));
}
__device__ __forceinline__ float softplus_f(float v) {
    return v > 20.0f ? v : log1pf(__expf(v));
}

// Stage `NROWS` rows of `KC` floats (16B per thread per round) into LDS.
// 128 threads; each round covers 512 floats = (512/KC) rows.
template <int KC, int NROWS>
__device__ __forceinline__ void stage_tile(const float* __restrict__ gbase,
                                           int grow_stride, float* lbase, int tid) {
    const int sn = tid / (KC / 4);          // row within round group
    const int sk = (tid % (KC / 4)) * 4;    // k offset (floats)
    constexpr int RPR = 512 / KC;           // rows per round
#pragma unroll
    for (int r = 0; r < NROWS / RPR; ++r) {
        const int row = r * RPR + sn;
        const float* gp = gbase + (size_t)row * grow_stride + sk;
        float* lp = lbase + row * KC + sk;
#if USE_ASYNC_LDS
        __builtin_amdgcn_global_load_async_to_lds_b128((gv4i_p)gp, (lv4i_p)lp, 0, 0);
#else
        const float4 v = *(const float4*)gp;
        *(float4*)lp = v;
#endif
    }
}

// ---------------------------------------------------------------------------
// fp32 WMMA GEMM: out[M x N] = A[M x K(lda)] * W[N x K]^T
// Block = 128 threads = 4 waves along M (64 rows); wave tile 16 x (16*NT).
// A (64 x KC) and B (16*NT x KC) chunks double-buffered in LDS via async
// global->LDS loads; fragments read with ds_load_b64; all WMMAs of a chunk
// issued back-to-back so load latency is overlapped across chunks.
// ---------------------------------------------------------------------------
template <int NT, int KC, int EPI>
__global__ __launch_bounds__(128)
void gemm_wmma_f32(const float* __restrict__ A, const float* __restrict__ W,
                   float* __restrict__ out0, float* __restrict__ out1,
                   const float* __restrict__ bias,
                   int M, int N, int K, int lda) {
    constexpr int BROWS = NT * 16;
    constexpr int TROWS = 64 + BROWS;
    constexpr int OPS   = (TROWS * KC) / 512;     // async ops per thread per chunk
    __shared__ float sm[2][TROWS * KC];

    const int tid   = threadIdx.x;
    const int lane  = tid & 31;
    const int wave  = tid >> 5;
    const int m0    = (blockIdx.x * 4 + wave) * 16;
    const int n0    = blockIdx.y * BROWS;
    const int khalf = 2 * (lane >> 4);            // 0 or 2
    const int ml    = wave * 16 + (lane & 15);    // A row within block tile
    const int nl    = lane & 15;                  // B row within 16-tile

    v8f acc[NT] = {};

    const float* gA0 = A + (size_t)(blockIdx.x * 64) * lda;
    const float* gB0 = W + (size_t)n0 * K;
    const int NCHK = K / KC;

    // prologue: stage chunk 0 into buffer 0
    stage_tile<KC, 64>(gA0, lda, &sm[0][0], tid);
    stage_tile<KC, BROWS>(gB0, K, &sm[0][64 * KC], tid);

    for (int c = 0; c < NCHK; ++c) {
        const int buf = c & 1;
        if (c + 1 < NCHK) {
            stage_tile<KC, 64>(gA0 + (c + 1) * KC, lda, &sm[buf ^ 1][0], tid);
            stage_tile<KC, BROWS>(gB0 + (c + 1) * KC, K, &sm[buf ^ 1][64 * KC], tid);
#if USE_ASYNC_LDS
            __builtin_amdgcn_s_wait_asynccnt(OPS);   // chunk c's loads complete
#endif
        } else {
#if USE_ASYNC_LDS
            __builtin_amdgcn_s_wait_asynccnt(0);
#endif
        }
        __syncthreads();

        const float* la = &sm[buf][ml * KC + khalf];
        const float* lb = &sm[buf][64 * KC + nl * KC + khalf];
        v2f a[KC / 4];
#pragma unroll
        for (int kk = 0; kk < KC / 4; ++kk)
            a[kk] = *(const v2f*)(la + kk * 4);
#pragma unroll
        for (int kk = 0; kk < KC / 4; ++kk) {
            v2f b[NT];
#pragma unroll
            for (int t = 0; t < NT; ++t)
                b[t] = *(const v2f*)(lb + t * 16 * KC + kk * 4);
#pragma unroll
            for (int t = 0; t < NT; ++t)
                acc[t] = __builtin_amdgcn_wmma_f32_16x16x4_f32(
                    false, a[kk], false, b[t], (short)0, acc[t], false, false);
        }
        __syncthreads();
    }

    // Epilogue (C/D layout: VGPR r -> row r + 8*(lane>>4), col = lane&15)
#pragma unroll
    for (int t = 0; t < NT; ++t) {
        const int col = n0 + t * 16 + nl;
        float bv = 0.0f;
        if (EPI == EP_SOFTPLUS) bv = bias[col];
#pragma unroll
        for (int r = 0; r < 8; ++r) {
            const int row = m0 + r + 8 * (lane >> 4);
            float v = acc[t][r];
            if (EPI == EP_SOFTPLUS) v = softplus_f(v + bv);
            if (EPI == EP_SPLIT) {
                if (col < DI) out0[(size_t)row * DI + col] = v;
                else          out1[(size_t)row * DI + (col - DI)] = v;
            } else {
                out0[(size_t)row * N + col] = v;
            }
        }
    }
}

// ---------------------------------------------------------------------------
// Causal depthwise conv1d (k=4, left pad 3) + bias + SiLU.
// ---------------------------------------------------------------------------
__global__ __launch_bounds__(256)
void conv_silu_kernel(const float* __restrict__ xs_raw,
                      const float* __restrict__ wconv,
                      const float* __restrict__ bconv,
                      float* __restrict__ xs) {
    const int idx = blockIdx.x * 256 + threadIdx.x;
    if (idx >= ROWS * DI) return;
    const int d    = idx % DI;
    const int row  = idx / DI;      // b*SL + l
    const int l    = row % SL;
    const int base = row - l;       // b*SL
    float acc = bconv[d];
#pragma unroll
    for (int k = 0; k < DC; ++k) {
        const int l2 = l - (DC - 1) + k;
        const float xv = (l2 >= 0) ? xs_raw[(size_t)(base + l2) * DI + d] : 0.0f;
        acc = fmaf(wconv[d * DC + k], xv, acc);
    }
    xs[idx] = silu_f(acc);
}

// ---------------------------------------------------------------------------
// Chunked parallel scan over the linear recurrence h_l = a_l h_{l-1} + b_l,
// a_l = exp(A*dt_l), b_l = dt_l*x_l*B_l.  16 chunks of 128 steps.
// gid mapping (passes 1 & 3): gid = ((b*CH + c)*DI + d)*DS + n.
// ---------------------------------------------------------------------------
__global__ __launch_bounds__(256)
void scan_pass1(const float* __restrict__ delta, const float* __restrict__ xs,
                const float* __restrict__ xdbl, const float* __restrict__ A_log,
                float* __restrict__ Aagg, float* __restrict__ Bagg) {
    const int gid = blockIdx.x * 256 + threadIdx.x;
    const int n   = gid & (DS - 1);
    const int t1  = gid >> 4;
    const int d   = t1 % DI;
    const int bc  = t1 / DI;
    const int c   = bc % CH;
    const int b   = bc / CH;

    const float a = -__expf(A_log[d * DS + n]);
    const int row0 = b * SL + c * CL;
    int idxD = row0 * DI + d;
    int idxB = row0 * XPN + DTR + n;

    float hB = 0.0f, ssum = 0.0f;
    float dt = delta[idxD], xt = xs[idxD], Bt = xdbl[idxB];
    for (int l = 0; l < CL; ++l) {
        const float dtc = dt, xtc = xt, Btc = Bt;
        if (l + 1 < CL) {                       // register-pipeline next step
            dt = delta[idxD + DI];
            xt = xs[idxD + DI];
            Bt = xdbl[idxB + XPN];
        }
        idxD += DI; idxB += XPN;
        const float al = __expf(a * dtc);
        hB = fmaf(al, hB, (dtc * xtc) * Btc);
        ssum += dtc;
    }
    Aagg[gid] = __expf(a * ssum);               // prod exp(a*dt) = exp(a*sum dt)
    Bagg[gid] = hB;
}

__global__ __launch_bounds__(256)
void scan_pass2(const float* __restrict__ Aagg, const float* __restrict__ Bagg,
                float* __restrict__ Hinit) {
    const int gid = blockIdx.x * 256 + threadIdx.x;   // BB*DI*DS
    const int n   = gid & (DS - 1);
    const int d   = (gid >> 4) % DI;
    const int b   = (gid >> 4) / DI;
    float H = 0.0f;
#pragma unroll
    for (int c = 0; c < CH; ++c) {
        const int idx = (((b * CH + c) * DI) + d) * DS + n;
        Hinit[idx] = H;
        H = fmaf(Aagg[idx], H, Bagg[idx]);
    }
}

__global__ __launch_bounds__(256)
void scan_pass3(const float* __restrict__ delta, const float* __restrict__ xs,
                const float* __restrict__ xdbl, const float* __restrict__ res,
                const float* __restrict__ A_log, const float* __restrict__ Dvec,
                const float* __restrict__ Hinit, float* __restrict__ ygate) {
    const int gid = blockIdx.x * 256 + threadIdx.x;
    const int n   = gid & (DS - 1);
    const int t1  = gid >> 4;
    const int d   = t1 % DI;
    const int bc  = t1 / DI;
    const int c   = bc % CH;
    const int b   = bc / CH;

    const float a  = -__expf(A_log[d * DS + n]);
    const float Dd = Dvec[d];
    const int row0 = b * SL + c * CL;
    int idxD = row0 * DI + d;
    int idxB = row0 * XPN + DTR + n;

    float h = Hinit[gid];
    float dt = delta[idxD], xt = xs[idxD];
    float Bt = xdbl[idxB], Ct = xdbl[idxB + DS];
    for (int l = 0; l < CL; ++l) {
        const float dtc = dt, xtc = xt, Btc = Bt, Ctc = Ct;
        if (l + 1 < CL) {
            dt = delta[idxD + DI];
            xt = xs[idxD + DI];
            Bt = xdbl[idxB + XPN];
            Ct = xdbl[idxB + XPN + DS];
        }
        const float al = __expf(a * dtc);
        h = fmaf(al, h, (dtc * xtc) * Btc);

        float yp = h * Ctc;                     // y_d = sum_n h*C_n
        yp += __shfl_xor(yp, 1);
        yp += __shfl_xor(yp, 2);
        yp += __shfl_xor(yp, 4);
        yp += __shfl_xor(yp, 8);

        if (n == 0) {
            const float rv = res[idxD];
            ygate[idxD] = (yp + xtc * Dd) * silu_f(rv);
        }
        idxD += DI; idxB += XPN;
    }
}

// ---------------------------------------------------------------------------
extern "C" void kernel_launch(void* const* d_in, const int* in_sizes, int n_in,
                              void* d_out, int out_size, void* d_ws, size_t ws_size,
                              hipStream_t stream) {
    const float* x       = (const float*)d_in[0];
    const float* W_in    = (const float*)d_in[1];
    const float* W_conv  = (const float*)d_in[2];
    const float* b_conv  = (const float*)d_in[3];
    const float* W_xproj = (const float*)d_in[4];
    const float* W_dt    = (const float*)d_in[5];
    const float* b_dt    = (const float*)d_in[6];
    const float* A_log   = (const float*)d_in[7];
    const float* Dv      = (const float*)d_in[8];
    const float* W_out   = (const float*)d_in[9];
    float* out = (float*)d_out;

    float* ws      = (float*)d_ws;
    float* xs_raw  = ws;                              // ROWS*DI
    float* res     = xs_raw + (size_t)ROWS * DI;      // ROWS*DI
    float* xs      = res    + (size_t)ROWS * DI;      // ROWS*DI
    float* xdbl    = xs     + (size_t)ROWS * DI;      // ROWS*XPN
    float* deltab  = xdbl   + (size_t)ROWS * XPN;     // ROWS*DI
    float* Aagg    = deltab + (size_t)ROWS * DI;      // BB*CH*DI*DS
    float* Bagg    = Aagg   + (size_t)BB * CH * DI * DS;
    float* Hinit   = Bagg   + (size_t)BB * CH * DI * DS;
    float* ygate   = xs_raw;                          // alias: xs_raw dead after conv

    const dim3 blk(128);

    // 1) in_proj: (4096 x 3072) = x @ W_in^T ; split xs_raw / res
    gemm_wmma_f32<4, 32, EP_SPLIT><<<dim3(ROWS / 64, (2 * DI) / 64), blk, 0, stream>>>(
        x, W_in, xs_raw, res, nullptr, ROWS, 2 * DI, DM, DM);

    // 2) causal depthwise conv + SiLU
    conv_silu_kernel<<<(ROWS * DI + 255) / 256, 256, 0, stream>>>(
        xs_raw, W_conv, b_conv, xs);

    // 3) x_proj: (4096 x 80) = xs @ W_xproj^T   (NT=5 covers all 80 cols)
    gemm_wmma_f32<5, 32, EP_PLAIN><<<dim3(ROWS / 64, 1), blk, 0, stream>>>(
        xs, W_xproj, xdbl, nullptr, nullptr, ROWS, XPN, DI, DI);

    // 4) dt_proj + softplus: (4096 x 1536) = xdbl[:, :48] @ W_dt^T + b_dt
    //    A slice has leading dim 80; K = 48 -> KC = 16 (3 chunks).
    gemm_wmma_f32<4, 16, EP_SOFTPLUS><<<dim3(ROWS / 64, DI / 64), blk, 0, stream>>>(
        xdbl, W_dt, deltab, nullptr, b_dt, ROWS, DI, DTR, XPN);

    // 5) chunked parallel selective scan + D-skip + gate
    scan_pass1<<<(BB * CH * DI * DS) / 256, 256, 0, stream>>>(
        deltab, xs, xdbl, A_log, Aagg, Bagg);
    scan_pass2<<<(BB * DI * DS) / 256, 256, 0, stream>>>(Aagg, Bagg, Hinit);
    scan_pass3<<<(BB * CH * DI * DS) / 256, 256, 0, stream>>>(
        deltab, xs, xdbl, res, A_log, Dv, Hinit, ygate);

    // 6) out_proj: (4096 x 768) = ygate @ W_out^T
    gemm_wmma_f32<4, 32, EP_PLAIN><<<dim3(ROWS / 64, DM / 64), blk, 0, stream>>>(
        ygate, W_out, out, nullptr, nullptr, ROWS, DM, DI, DI);
}